// TestingModule_82282983457187
// MI455X (gfx1250) — compile-verified
//
#include <hip/hip_runtime.h>
#include <cstdint>

// Soft-NMS for MI455X (gfx1250).
// Phase 1 (whole chip): precompute decay matrix W[j][i] = exp(-2*iou(j,i)^2)  (64MB, L2-resident)
// Phase 2 (one WGP):    sequential selection loop, all state in registers/LDS,
//                       2 barriers per iteration, row of W streamed from L2 with NT hint.
// CDNA5 paths: global_load_async_to_lds_b128 + s_wait_asynccnt (async DMA to LDS),
//              wave32 shuffle butterflies (ds path), split s_barrier ops, s_wait_* counters.

#define SNMS_T   1024   // threads in sequential kernel (32 wave32 waves, one WGP)
#define SNMS_EPT 4      // elements per thread for N=4096

__device__ __forceinline__ uint32_t lds_off_u32(const void* p) {
  // Generic (flat) pointer to LDS: low 32 bits are the LDS byte offset
  // (aperture rule in CDNA5 ISA).
  return (uint32_t)(uintptr_t)p;
}

__device__ __forceinline__ void async_copy_b128_to_lds(uint32_t lds_byte_addr,
                                                       const void* gptr) {
  // ASYNCcnt-tracked DMA of 16 bytes per lane from global memory into LDS
  asm volatile("global_load_async_to_lds_b128 %0, %1, off"
               :
               : "v"(lds_byte_addr), "v"(gptr)
               : "memory");
}

__device__ __forceinline__ void wait_asynccnt0() {
  asm volatile("s_wait_asynccnt 0" ::: "memory");
}

__device__ __forceinline__ float iou_weight(float4 a, float area_a,
                                            float4 b, float area_b) {
  float xx1 = fmaxf(a.x, b.x);
  float yy1 = fmaxf(a.y, b.y);
  float xx2 = fminf(a.z, b.z);
  float yy2 = fminf(a.w, b.w);
  float w = fmaxf(xx2 - xx1, 0.0f);
  float h = fmaxf(yy2 - yy1, 0.0f);
  float inter = w * h;
  float iou = inter / (area_a + area_b - inter);
  // weight = exp(-(iou^2)/sigma), sigma = 0.5  ->  exp(-2*iou^2)
  return __expf(-2.0f * iou * iou);
}

// ---------------------------------------------------------------------------
// Phase 1: W[j*n + i] = exp(-2*iou(box_j, box_i)^2). grid = (n/256, n).
// Each lane async-DMAs its own box into LDS (CDNA5 async-to-LDS path).
// ---------------------------------------------------------------------------
__global__ __launch_bounds__(256) void softnms_weights_kernel(
    const float4* __restrict__ boxes, float* __restrict__ W, int n) {
  __shared__ float4 sb[256];
  const int tid = threadIdx.x;
  const int i = blockIdx.x * 256 + tid;
  const int j = blockIdx.y;

  async_copy_b128_to_lds(lds_off_u32(&sb[tid]), &boxes[i]);

  // Uniform address -> scalar load of the selected box, overlapped with DMA.
  const float4 bj = boxes[j];
  const float aj = (bj.z - bj.x) * (bj.w - bj.y);

  wait_asynccnt0();                 // own wave's async copies done
  const float4 bi = sb[tid];        // each lane reads only what it wrote
  const float ai = (bi.z - bi.x) * (bi.w - bi.y);

  // Regular (RT) store: keep the row resident in the 192MB L2 for phase 2.
  W[(size_t)j * (size_t)n + (size_t)i] = iou_weight(bj, aj, bi, ai);
}

// ---------------------------------------------------------------------------
// Phase 2: sequential Soft-NMS loop on a single workgroup.
// Per-thread registers hold 4 strided scores/processed-flags (+ boxes/areas in
// the fused instantiation). LDS holds a box mirror (random access by j) +
// argmax reduction slots.
// ---------------------------------------------------------------------------
template <int USE_W>
__global__ __launch_bounds__(SNMS_T) void softnms_seq_kernel(
    const float4* __restrict__ boxes, const float* __restrict__ scores,
    const float* __restrict__ W, float* __restrict__ out, int n) {
  __shared__ float4 s_box[SNMS_T * SNMS_EPT];  // 64 KB of the WGP's 320 KB LDS
  __shared__ float  red_v[SNMS_T / 32];
  __shared__ int    red_i[SNMS_T / 32];
  __shared__ int    s_j;

  const int tid  = threadIdx.x;
  const int lane = tid & 31;
  const int wid  = tid >> 5;

  float  sc[SNMS_EPT];
  float4 bx[SNMS_EPT];
  float  ar[SNMS_EPT];
  bool   done[SNMS_EPT];

  // Stage boxes into LDS via async DMA, then mirror into registers.
#pragma unroll
  for (int k = 0; k < SNMS_EPT; ++k) {
    const int i = tid + k * SNMS_T;
    async_copy_b128_to_lds(lds_off_u32(&s_box[i]), &boxes[i]);
  }
  wait_asynccnt0();
#pragma unroll
  for (int k = 0; k < SNMS_EPT; ++k) {
    const int i = tid + k * SNMS_T;
    const float4 b = s_box[i];      // own lane's DMA result
    bx[k]   = b;                    // dead (DCE'd) in the USE_W instantiation
    ar[k]   = (b.z - b.x) * (b.w - b.y);
    sc[k]   = scores[i];
    done[k] = false;
  }
  __syncthreads();  // LDS box mirror visible to everyone

  for (int it = 0; it < n; ++it) {
    // ---- local argmax over this thread's unprocessed elements ----
    float bv = -__builtin_huge_valf();
    int   bi = 0x7fffffff;
#pragma unroll
    for (int k = 0; k < SNMS_EPT; ++k) {
      const int i = tid + k * SNMS_T;
      if (!done[k] && sc[k] > bv) { bv = sc[k]; bi = i; }  // ascending scan -> lowest idx on tie
    }
    // ---- wave32 butterfly reduction (value, lowest index on ties) ----
#pragma unroll
    for (int off = 16; off > 0; off >>= 1) {
      const float ov = __shfl_xor(bv, off, 32);
      const int   oi = __shfl_xor(bi, off, 32);
      if (ov > bv || (ov == bv && oi < bi)) { bv = ov; bi = oi; }
    }
    if (lane == 0) { red_v[wid] = bv; red_i[wid] = bi; }
    __syncthreads();  // barrier 1: partials ready, old s_j fully consumed

    if (wid == 0) {
      float v  = red_v[lane];
      int   ix = red_i[lane];
#pragma unroll
      for (int off = 16; off > 0; off >>= 1) {
        const float ov = __shfl_xor(v, off, 32);
        const int   oi = __shfl_xor(ix, off, 32);
        if (ov > v || (ov == v && oi < ix)) { v = ov; ix = oi; }
      }
      if (lane == 0) s_j = ix;
    }
    __syncthreads();  // barrier 2: selected index published

    const int j = s_j;

    // Critical-path latency hiding: issue ALL row loads immediately and
    // unconditionally (row is always fully valid), non-temporal since every
    // W element is consumed exactly once ever (hits L2, skips near-cache
    // retention). Conditions applied afterwards on registers only.
    float wrow[SNMS_EPT];
    if (USE_W) {
      const float* __restrict__ row = W + (size_t)j * (size_t)n;
#pragma unroll
      for (int k = 0; k < SNMS_EPT; ++k) {
        wrow[k] = __builtin_nontemporal_load(row + (tid + k * SNMS_T));
      }
    }

    float4 bj; float aj = 0.0f;
    if (!USE_W) {                    // box[j] only needed to compute IoU
      bj = s_box[j];
      aj = (bj.z - bj.x) * (bj.w - bj.y);
    }

#pragma unroll
    for (int k = 0; k < SNMS_EPT; ++k) {
      const int i = tid + k * SNMS_T;
      if (done[k]) continue;
      if (i == j) {
        // Score frozen at selection time; emit score + keep flag exactly once.
        out[j]     = sc[k];
        out[n + j] = (sc[k] > 0.05f) ? 1.0f : 0.0f;
        done[k]    = true;
      } else {
        const float w = USE_W ? wrow[k] : iou_weight(bj, aj, bx[k], ar[k]);
        sc[k] *= w;
      }
    }
    // No barrier needed here: next iteration only touches registers before
    // barrier 1, and red_/s_j rewrites are already fenced by the two barriers.
  }
}

extern "C" void kernel_launch(void* const* d_in, const int* in_sizes, int n_in,
                              void* d_out, int out_size, void* d_ws, size_t ws_size,
                              hipStream_t stream) {
  (void)n_in; (void)out_size;
  const float4* boxes  = (const float4*)d_in[0];
  const float*  scores = (const float*)d_in[1];
  float*        out    = (float*)d_out;
  const int     n      = in_sizes[1];  // 4096; SNMS_T*SNMS_EPT must equal n

  const size_t need = (size_t)n * (size_t)n * sizeof(float);
  if (ws_size >= need) {
    float* W = (float*)d_ws;
    softnms_weights_kernel<<<dim3((unsigned)(n / 256), (unsigned)n), 256, 0, stream>>>(
        boxes, W, n);
    softnms_seq_kernel<1><<<1, SNMS_T, 0, stream>>>(boxes, scores, W, out, n);
  } else {
    softnms_seq_kernel<0><<<1, SNMS_T, 0, stream>>>(boxes, scores, nullptr, out, n);
  }
}